// GNNModel_420906795156
// MI455X (gfx1250) — compile-verified
//
#include <hip/hip_runtime.h>
#include <stdint.h>

// ---- problem constants (from reference) ----
#define N_NODES 50000
#define N_EDGES 800000
#define C_IN    128
#define C_HID   128
#define C_OUT   64

typedef float v2f __attribute__((ext_vector_type(2)));
typedef float v8f __attribute__((ext_vector_type(8)));

// ---------------- utility kernels ----------------
__global__ void fill_kernel(float* __restrict__ p, float v, int n) {
    int i = blockIdx.x * blockDim.x + threadIdx.x;
    if (i < n) p[i] = v;
}

__global__ void zero_kernel(float* __restrict__ p, long long n) {
    long long i = (long long)blockIdx.x * blockDim.x + threadIdx.x;
    if (i < n) p[i] = 0.0f;
}

// deg[dst] += 1 per edge (self-loop pre-seeded as 1.0 by fill_kernel)
__global__ void degree_kernel(const long long* __restrict__ dst, float* __restrict__ deg, int nE) {
    int e = blockIdx.x * blockDim.x + threadIdx.x;
    if (e < nE) atomicAdd(&deg[(int)dst[e]], 1.0f);
}

// dinv = rsqrt(deg); deg >= 1 always (self loops), matching reference's >0 guard
__global__ void rsqrt_kernel(float* __restrict__ p, int n) {
    int i = blockIdx.x * blockDim.x + threadIdx.x;
    if (i < n) p[i] = rsqrtf(p[i]);
}

// ---------------- fp32 WMMA GEMM: out[N, NCOL] = A[N, K] * W[K, NCOL] ----------------
// One wave computes one 16x16 C tile; blockDim.x = 32 * (NCOL/16) waves cover all
// column tiles of one 16-row stripe. N_NODES is an exact multiple of 16.
//
// V_WMMA_F32_16X16X4_F32 operand layouts (ISA 7.12.2, wave32):
//   A (16x4, MxK): lanes 0-15 -> M=lane, {v0,v1}={K0,K1}; lanes 16-31 -> {K2,K3}
//   B (4x16, KxN): lanes 0-15 -> N=lane, {v0,v1}={K0,K1}; lanes 16-31 -> {K2,K3}
//   C/D (16x16)  : VGPR r -> M = r + 8*(lane>>4), N = lane&15
template <int K, int NCOL>
__global__ void gemm_wmma_f32(const float* __restrict__ A,
                              const float* __restrict__ W,
                              float* __restrict__ out) {
    const int lane = threadIdx.x & 31;
    const int wave = threadIdx.x >> 5;
    const int row0 = blockIdx.x << 4;   // 16-row stripe
    const int col0 = wave << 4;         // 16-col tile per wave
    const int half = lane >> 4;         // 0: K pair {0,1}, 1: K pair {2,3}
    const int l15  = lane & 15;

    const float* aRow = A + (size_t)(row0 + l15) * K;
    const float* wCol = W + col0 + l15;

    v8f acc = {0.f, 0.f, 0.f, 0.f, 0.f, 0.f, 0.f, 0.f};

#pragma unroll 4
    for (int k0 = 0; k0 < K; k0 += 4) {
        const int kk = k0 + half * 2;
        v2f a;
        a.x = aRow[kk];
        a.y = aRow[kk + 1];
        v2f b;
        b.x = wCol[(size_t)kk * NCOL];
        b.y = wCol[(size_t)(kk + 1) * NCOL];
        acc = __builtin_amdgcn_wmma_f32_16x16x4_f32(
            /*neg_a=*/false, a, /*neg_b=*/false, b,
            /*c_mod=*/(short)0, acc, /*reuse_a=*/false, /*reuse_b=*/false);
    }

    float* o = out + (size_t)row0 * NCOL + col0 + l15;
#pragma unroll
    for (int r = 0; r < 8; ++r) {
        const int m = r + half * 8;
        o[(size_t)m * NCOL] = acc[r];
    }
}

// ---------------- edge scatter: agg[dst] += h[src] * dinv[src]*dinv[dst] ----------------
// One wave32 per edge; each lane owns C/32 contiguous channels (float4/float2 gather,
// per-float L2 atomics). h fits in the 192MB L2, so gathers + atomics stay on-chip.
template <int C>
__global__ void scatter_kernel(const float* __restrict__ h,
                               const long long* __restrict__ src,
                               const long long* __restrict__ dst,
                               const float* __restrict__ dinv,
                               float* __restrict__ agg, int nE) {
    const long long gid = (long long)blockIdx.x * blockDim.x + threadIdx.x;
    const int e    = (int)(gid >> 5);
    const int lane = (int)(gid & 31);
    if (e >= nE) return;

    const int s = (int)src[e];
    const int d = (int)dst[e];
    const float w = dinv[s] * dinv[d];

    constexpr int VEC = C / 32;  // 4 for C=128, 2 for C=64
    const float* hs = h + (size_t)s * C + lane * VEC;
    float* ad = agg + (size_t)d * C + lane * VEC;

    if constexpr (VEC == 4) {
        const float4 v = *reinterpret_cast<const float4*>(hs);
        atomicAdd(ad + 0, v.x * w);
        atomicAdd(ad + 1, v.y * w);
        atomicAdd(ad + 2, v.z * w);
        atomicAdd(ad + 3, v.w * w);
    } else {
        const float2 v = *reinterpret_cast<const float2*>(hs);
        atomicAdd(ad + 0, v.x * w);
        atomicAdd(ad + 1, v.y * w);
    }
}

// ---------------- self-loop + bias (+ optional ReLU), in place on agg ----------------
template <int C, bool RELU>
__global__ void finalize_kernel(float* __restrict__ agg,
                                const float* __restrict__ h,
                                const float* __restrict__ dinv,
                                const float* __restrict__ bias,
                                long long total) {
    long long i = (long long)blockIdx.x * blockDim.x + threadIdx.x;
    if (i >= total) return;
    const int node = (int)(i / C);   // C is a power of two -> shift
    const int c    = (int)(i % C);
    const float di = dinv[node];
    float v = agg[i] + h[i] * (di * di) + bias[c];
    agg[i] = RELU ? fmaxf(v, 0.0f) : v;
}

// ---------------- launch ----------------
extern "C" void kernel_launch(void* const* d_in, const int* in_sizes, int n_in,
                              void* d_out, int out_size, void* d_ws, size_t ws_size,
                              hipStream_t stream) {
    (void)in_sizes; (void)n_in; (void)out_size; (void)ws_size;

    const float*     x  = (const float*)d_in[0];
    const long long* ei = (const long long*)d_in[1];  // [2, N_EDGES], int64 per reference
    const float*     W1 = (const float*)d_in[2];
    const float*     b1 = (const float*)d_in[3];
    const float*     W2 = (const float*)d_in[4];
    const float*     b2 = (const float*)d_in[5];
    const long long* src = ei;
    const long long* dst = ei + N_EDGES;
    float* outp = (float*)d_out;

    // scratch layout (~64.3 MB): dinv | h1 | agg1 | h2
    char* ws = (char*)d_ws;
    size_t off = 0;
    auto alloc = [&](size_t bytes) -> void* {
        off = (off + 255) & ~(size_t)255;
        void* p = ws + off;
        off += bytes;
        return p;
    };
    float* dinv = (float*)alloc((size_t)N_NODES * sizeof(float));
    float* h1   = (float*)alloc((size_t)N_NODES * C_HID * sizeof(float));
    float* agg1 = (float*)alloc((size_t)N_NODES * C_HID * sizeof(float));
    float* h2   = (float*)alloc((size_t)N_NODES * C_OUT * sizeof(float));

    const int BT = 256;
    const unsigned nodeBlocks  = (N_NODES + BT - 1) / BT;
    const unsigned edgeBlocks  = (N_EDGES + BT - 1) / BT;
    const unsigned edgeWaveBlk = (unsigned)(((long long)N_EDGES * 32 + BT - 1) / BT); // 100000
    const long long t1 = (long long)N_NODES * C_HID;
    const long long t2 = (long long)N_NODES * C_OUT;
    const unsigned t1Blocks = (unsigned)((t1 + BT - 1) / BT);
    const unsigned t2Blocks = (unsigned)((t2 + BT - 1) / BT);

    // degrees (self-loop seeded as 1.0) -> dinv = rsqrt(deg)
    fill_kernel<<<nodeBlocks, BT, 0, stream>>>(dinv, 1.0f, N_NODES);
    degree_kernel<<<edgeBlocks, BT, 0, stream>>>(dst, dinv, N_EDGES);
    rsqrt_kernel<<<nodeBlocks, BT, 0, stream>>>(dinv, N_NODES);

    // ---- layer 1: h1 = x @ W1 ; agg1 = norm-scatter(h1) ; relu(agg1 + selfloop + b1)
    gemm_wmma_f32<C_IN, C_HID><<<N_NODES / 16, 32 * (C_HID / 16), 0, stream>>>(x, W1, h1);
    zero_kernel<<<t1Blocks, BT, 0, stream>>>(agg1, t1);
    scatter_kernel<C_HID><<<edgeWaveBlk, BT, 0, stream>>>(h1, src, dst, dinv, agg1, N_EDGES);
    finalize_kernel<C_HID, true><<<t1Blocks, BT, 0, stream>>>(agg1, h1, dinv, b1, t1);

    // ---- layer 2: h2 = agg1 @ W2 ; out = norm-scatter(h2) + selfloop + b2
    gemm_wmma_f32<C_HID, C_OUT><<<N_NODES / 16, 32 * (C_OUT / 16), 0, stream>>>(agg1, W2, h2);
    zero_kernel<<<t2Blocks, BT, 0, stream>>>(outp, t2);
    scatter_kernel<C_OUT><<<edgeWaveBlk, BT, 0, stream>>>(h2, src, dst, dinv, outp, N_EDGES);
    finalize_kernel<C_OUT, false><<<t2Blocks, BT, 0, stream>>>(outp, h2, dinv, b2, t2);
}